// AtomAttention_12876311954009
// MI455X (gfx1250) — compile-verified
//
#include <hip/hip_runtime.h>
#include <hip/hip_bf16.h>

// AtomAttention on MI455X (gfx1250), wave32 + WMMA fp32.
//
// Key algebraic simplification: relu((c_i+c_j)@Wa_pair) = relu(P_i+P_j) with
// P = c_batch @ Wa_pair, so the dominant [64^3 pairs]x[64x256] GEMM collapses
// to one [64x64]@[64x256] GEMM per molecule. One workgroup per molecule,
// everything staged in LDS, all three GEMMs on v_wmma_f32_16x16x4_f32.

typedef __attribute__((ext_vector_type(2))) float v2f;
typedef __attribute__((ext_vector_type(4))) float v4f;
typedef __attribute__((ext_vector_type(8))) float v8f;

#define BSZ 64
#define MA 64
#define HD 64
#define NHD 4
#define HP 256   // HD*NHD

// LDS layout in floats (144 KB total, dynamic):
//   [0,      4096)  sC   : c_batch 64x64        (reused as sOut after GEMM2)
//   [4096,  20480)  sP   : P 64x256             (reused as c_sum interleaved)
//   [20480, 36864)  sAtt : att scores 4 x 64x64 (one plane per head)
#define OFF_C   0
#define OFF_P   4096
#define OFF_ATT 20480
#define LDS_FLOATS 36864

// A fragment: 16x4 fp32 tile of row-major src (leading dim ld) at (row0, k0).
// Lanes 0-15: M=lane, K=k0+{0,1}; lanes 16-31: M=lane-16, K=k0+{2,3}.
__device__ __forceinline__ v2f frag_a(const float* __restrict__ src, int ld,
                                      int row0, int k0, int lane) {
  int m  = lane & 15;
  int kk = (lane >> 4) << 1;
  const float* p = src + (row0 + m) * ld + k0 + kk;
  v2f a; a.x = p[0]; a.y = p[1];
  return a;
}

// B fragment: 4x16 fp32 tile of row-major [K x N] src at (k0, col0).
// Lanes 0-15: N=lane, K=k0+{0,1}; lanes 16-31: N=lane-16, K=k0+{2,3}.
__device__ __forceinline__ v2f frag_b(const float* __restrict__ src, int ld,
                                      int k0, int col0, int lane) {
  int n  = lane & 15;
  int kk = (lane >> 4) << 1;
  v2f b;
  b.x = src[(k0 + kk) * ld + col0 + n];
  b.y = src[(k0 + kk + 1) * ld + col0 + n];
  return b;
}

__global__ void __launch_bounds__(256)
atom_attention_kernel(const float* __restrict__ inputs,     // [N+1, 64]
                      const int*   __restrict__ scope,      // [64, 64]
                      const float* __restrict__ Wa_pair,    // [64, 256]
                      const float* __restrict__ Wa_score,   // [64, 4]
                      const float* __restrict__ W_proj,     // [256, 64]
                      float* __restrict__ out_mol,          // [64, 64]
                      float* __restrict__ out_flat) {       // [N+1, 64]
  extern __shared__ float smem[];
  float* sC   = smem + OFF_C;
  float* sP   = smem + OFF_P;
  float* sAtt = smem + OFF_ATT;
  float* sS   = smem + OFF_P;   // c_sum interleaved [i][h*4+n], reuses sP
  float* sOut = smem + OFF_C;   // c_atom_att [i][h], reuses sC

  const int b    = blockIdx.x;
  const int tid  = threadIdx.x;
  const int lane = tid & 31;
  const int wave = tid >> 5;    // 0..7

  // ---- Phase 0: gather c_batch[b] = inputs[scope[b,:]] into LDS ------------
  for (int e = tid; e < MA * HD; e += 256) {
    int i = e >> 6, h = e & 63;
    int s = scope[b * MA + i];                 // 0 -> zero pad row of inputs
    sC[e] = inputs[s * HD + h];
  }
  __syncthreads();

  // ---- Phase 1: P = sC[64x64] @ Wa_pair[64x256]  (WMMA fp32) ---------------
  for (int t = wave; t < 64; t += 8) {         // 4x16 grid of 16x16 tiles
    int row0 = (t >> 4) << 4;
    int col0 = (t & 15) << 4;
    v8f acc = {};
#pragma unroll
    for (int k = 0; k < HD; k += 4) {
      v2f a  = frag_a(sC, HD, row0, k, lane);
      v2f bb = frag_b(Wa_pair, HP, k, col0, lane);
      acc = __builtin_amdgcn_wmma_f32_16x16x4_f32(
          false, a, false, bb, (short)0, acc, false, false);
    }
    int nn = lane & 15;
    int mb = (lane >> 4) << 3;
#pragma unroll
    for (int r = 0; r < 8; ++r)
      sP[(row0 + mb + r) * HP + col0 + nn] = acc[r];
  }
  __syncthreads();

  // ---- Phase 2: att[i,j,n] = sigmoid(sum_h relu(Pi+Pj)[4h+n] * Ws[h,n]) ----
  {
    int i     = tid >> 2;                      // fixed row per thread
    int jbase = (tid & 3) << 4;                // 16 consecutive j per thread
    const v4f* Pi = (const v4f*)(sP + i * HP);
    const v4f* Ws = (const v4f*)Wa_score;      // [h][n], 16B aligned rows
    for (int p = 0; p < 16; ++p) {
      int j = jbase + p;
      const v4f* Pj = (const v4f*)(sP + j * HP);
      v4f acc = {};
      for (int h = 0; h < HD; ++h) {
        v4f s = Pi[h] + Pj[h];
        s.x = fmaxf(s.x, 0.0f);
        s.y = fmaxf(s.y, 0.0f);
        s.z = fmaxf(s.z, 0.0f);
        s.w = fmaxf(s.w, 0.0f);
        acc += s * Ws[h];
      }
#pragma unroll
      for (int n = 0; n < NHD; ++n) {
        float v = acc[n];
        sAtt[n * (MA * MA) + i * MA + j] = 1.0f / (1.0f + __expf(-v));
      }
    }
  }
  __syncthreads();

  // ---- Phase 3: per head n: S_n = att_n[64x64] @ sC[64x64]  (WMMA) ---------
  // Store interleaved: sS[i][h*4+n]  (matches torch .view(h*nh) h-major).
  for (int g = wave; g < 64; g += 8) {         // 4 heads x (4x4 tiles)
    int head = g >> 4;
    int t    = g & 15;
    int row0 = (t >> 2) << 4;
    int col0 = (t & 3) << 4;
    const float* A = sAtt + head * (MA * MA);
    v8f acc = {};
#pragma unroll
    for (int k = 0; k < MA; k += 4) {
      v2f a  = frag_a(A, MA, row0, k, lane);
      v2f bb = frag_b(sC, HD, k, col0, lane);
      acc = __builtin_amdgcn_wmma_f32_16x16x4_f32(
          false, a, false, bb, (short)0, acc, false, false);
    }
    int nn = lane & 15;
    int mb = (lane >> 4) << 3;
#pragma unroll
    for (int r = 0; r < 8; ++r)
      sS[(row0 + mb + r) * HP + (col0 + nn) * NHD + head] = acc[r];
  }
  __syncthreads();

  // ---- Phase 4: c_atom_att = sS[64x256] @ W_proj[256x64]  (WMMA) -----------
  for (int t = wave; t < 16; t += 8) {         // 4x4 grid of 16x16 tiles
    int row0 = (t >> 2) << 4;
    int col0 = (t & 3) << 4;
    v8f acc = {};
#pragma unroll
    for (int k = 0; k < HP; k += 4) {
      v2f a  = frag_a(sS, HP, row0, k, lane);
      v2f bb = frag_b(W_proj, HD, k, col0, lane);
      acc = __builtin_amdgcn_wmma_f32_16x16x4_f32(
          false, a, false, bb, (short)0, acc, false, false);
    }
    int nn = lane & 15;
    int mb = (lane >> 4) << 3;
#pragma unroll
    for (int r = 0; r < 8; ++r)
      sOut[(row0 + mb + r) * HD + col0 + nn] = acc[r];
  }
  __syncthreads();

  // ---- Phase 5: molecule sum + ragged scatter ------------------------------
  if (tid < HD) {
    float s = 0.0f;
    for (int i = 0; i < MA; ++i) s += sOut[i * HD + tid];
    out_mol[b * HD + tid] = s;
  }
  // flat[scope[b,i]] = c_atom_att[b,i]; pad slots (scope==0) are skipped.
  for (int e = tid; e < MA * HD; e += 256) {
    int i = e >> 6, h = e & 63;
    int s = scope[b * MA + i];
    if (s != 0) out_flat[s * HD + h] = sOut[e];
  }
  if (b == 0 && tid < HD) out_flat[tid] = 0.0f;  // zero pad row 0
}

extern "C" void kernel_launch(void* const* d_in, const int* in_sizes, int n_in,
                              void* d_out, int out_size, void* d_ws, size_t ws_size,
                              hipStream_t stream) {
  (void)in_sizes; (void)n_in; (void)d_ws; (void)ws_size; (void)out_size;
  const float* inputs   = (const float*)d_in[0];
  const int*   scope    = (const int*)d_in[1];   // JAX default: int32
  // d_in[2] = scope_rev (unused; scatter uses scope directly)
  const float* Wa_pair  = (const float*)d_in[3];
  const float* Wa_score = (const float*)d_in[4];
  const float* W_proj   = (const float*)d_in[5];

  float* out_mol  = (float*)d_out;           // [64,64]
  float* out_flat = (float*)d_out + BSZ * HD; // [N+1,64]

  dim3 grid(BSZ), block(256);
  size_t shmem = (size_t)LDS_FLOATS * sizeof(float);  // 144 KB of the 320 KB WGP LDS
  atom_attention_kernel<<<grid, block, shmem, stream>>>(
      inputs, scope, Wa_pair, Wa_score, W_proj, out_mol, out_flat);
}